// TransformerSimple_85255100825617
// MI455X (gfx1250) — compile-verified
//
#include <hip/hip_runtime.h>
#include <hip/hip_bf16.h>
#include <math.h>

// ---------------------------------------------------------------------------
// MI455X (gfx1250) point-transformer. All dense math via V_WMMA_F32_16X16X4_F32
// (fp32 matrix path, matches the fp32 reference). Each wave32 owns a 32x64
// output strip (2 M-tiles x 4 N-tiles): per K-step, 2 A fragments + 4 shared
// B fragments feed 8 WMMAs. All operand addressing is integer offsets off the
// kernel-arg pointers so loads lower to global_load_* (not flat_*), and all
// loads of a K-step are grouped before the WMMA burst so one s_wait covers 8
// matrix ops. 192MB L2 keeps the 33MB/layer activation working set resident.
// ---------------------------------------------------------------------------

#define NP   65536   // points
#define INW  32      // in_dim
#define DH   128     // dim per head
#define NH   8       // heads
#define BLKA 128     // attention block size
#define KQ   132     // padded q/k width (128 + 3 coords + 1 pad)

typedef __attribute__((ext_vector_type(2))) float v2f;
typedef __attribute__((ext_vector_type(8))) float v8f;

enum {
  EPI_NONE = 0, EPI_RELU = 1, EPI_TANH = 2, EPI_EXPSCORE = 3,
  EPI_SCALE_ROW = 4, EPI_RES = 5, EPI_ACC = 6
};

// AMODE: 0 = A row-major [M x K]; 1 = A is [X0|X1|X2] concat along K (3x128).
// BT:    0 = B stored [K x N]; 1 = B stored [N x K] (Q @ K^T).
// BDIAG: 1 = block-local attention: B row index offset by (m0 & ~127).
template<int AMODE, int BT, int BDIAG, int EPI>
__global__ __launch_bounds__(128)
void gemm_wmma(const float* __restrict__ A,
               const float* __restrict__ A1,
               const float* __restrict__ A2,
               int lda,
               const float* __restrict__ B, int ldb,
               const float* __restrict__ bias,
               float* __restrict__ C, int ldc,
               int K,
               const float* __restrict__ rowterm,
               const float* __restrict__ colterm,
               const float* __restrict__ res, int ldres)
{
  const int lane = threadIdx.x & 31;
  const int wave = threadIdx.x >> 5;
  const int g    = lane >> 4;              // half-wave: K-pair select
  const int mr   = lane & 15;
  const int m0   = (blockIdx.y * 4 + wave) * 32;   // 2 M-tiles: m0, m0+16
  const int n0   = blockIdx.x * 64;                // 4 N-tiles
  const int bb   = BDIAG ? (m0 & ~(BLKA - 1)) : 0;

  v8f acc[2][4] = {};

  // Integer-offset addressing off the kernel-arg pointers (keeps global AS).
  size_t aoff0 = (size_t)(m0 + mr) * (AMODE ? 128 : lda) + 2 * g;
  size_t aoff1 = aoff0 + (size_t)16 * (AMODE ? 128 : lda);
  size_t boff  = BT ? ((size_t)(bb + n0 + mr) * ldb + 2 * g)
                    : ((size_t)(bb + 2 * g) * ldb + (n0 + mr));
  const size_t bts  = BT ? (size_t)16 * ldb : 16;   // N-tile stride in B
  const size_t bsec = BT ? 1 : (size_t)ldb;         // second-K-element stride
  const size_t badv = BT ? 4 : (size_t)4 * ldb;     // K-step advance

  const int nseg  = AMODE ? 3 : 1;
  const int kseg  = AMODE ? 128 : K;

  for (int seg = 0; seg < nseg; ++seg) {
    const float* __restrict__ As =
        (AMODE == 0) ? A : ((seg == 0) ? A : ((seg == 1) ? A1 : A2));
    size_t a0 = aoff0, a1 = aoff1;
    for (int k = 0; k < kseg; k += 4) {
      // -- load phase: 2 A fragments (contiguous pairs) + 4 B fragments --
      v2f av0, av1;
      av0.x = As[a0]; av0.y = As[a0 + 1];
      av1.x = As[a1]; av1.y = As[a1 + 1];
      a0 += 4; a1 += 4;
      v2f bv[4];
      #pragma unroll
      for (int t = 0; t < 4; ++t) {
        bv[t].x = B[boff + t * bts];
        bv[t].y = B[boff + t * bts + bsec];
      }
      boff += badv;
      // -- matrix phase: 8 WMMAs sharing the loaded fragments --
      #pragma unroll
      for (int t = 0; t < 4; ++t)
        acc[0][t] = __builtin_amdgcn_wmma_f32_16x16x4_f32(
            false, av0, false, bv[t], (short)0, acc[0][t], false, false);
      #pragma unroll
      for (int t = 0; t < 4; ++t)
        acc[1][t] = __builtin_amdgcn_wmma_f32_16x16x4_f32(
            false, av1, false, bv[t], (short)0, acc[1][t], false, false);
    }
  }

  // C/D layout: VGPR r holds row tile_m0 + r + 8*(lane>>4), col base + (lane&15).
  #pragma unroll
  for (int mt = 0; mt < 2; ++mt) {
    #pragma unroll
    for (int t = 0; t < 4; ++t) {
      #pragma unroll
      for (int r = 0; r < 8; ++r) {
        const int m  = m0 + mt * 16 + r + 8 * g;
        const int nc = n0 + 16 * t + mr;
        float val = acc[mt][t][r];
        if (bias) val += bias[nc];
        if      (EPI == EPI_RELU)      val = fmaxf(val, 0.0f);
        else if (EPI == EPI_TANH)      val = tanhf(val);
        else if (EPI == EPI_EXPSCORE)  val = __expf(fminf(val + rowterm[m] + colterm[bb + nc], 0.0f));
        else if (EPI == EPI_SCALE_ROW) val *= rowterm[m];
        else if (EPI == EPI_RES)       val += res[(size_t)m * ldres + nc];
        else if (EPI == EPI_ACC)       val += C[(size_t)m * ldc + nc];
        C[(size_t)m * ldc + nc] = val;
      }
    }
  }
}

// LayerNorm over rows of width W (128 or 256); one wave per row; optional tanh.
__global__ __launch_bounds__(256)
void layernorm_k(const float* __restrict__ x, const float* __restrict__ gam,
                 const float* __restrict__ bet, float* __restrict__ y,
                 int W, int do_tanh)
{
  const int row  = blockIdx.x * 8 + (threadIdx.x >> 5);
  const int lane = threadIdx.x & 31;
  const int per  = W >> 5;            // 4 or 8 elements per lane
  const float* xr = x + (size_t)row * W;
  float v[8];
  float s = 0.f;
  for (int i = 0; i < per; ++i) { v[i] = xr[lane + (i << 5)]; s += v[i]; }
  for (int off = 16; off; off >>= 1) s += __shfl_xor(s, off, 32);
  const float mu = s / W;
  float s2 = 0.f;
  for (int i = 0; i < per; ++i) { float d = v[i] - mu; s2 += d * d; }
  for (int off = 16; off; off >>= 1) s2 += __shfl_xor(s2, off, 32);
  const float inv = rsqrtf(s2 / W + 1e-5f);
  float* yr = y + (size_t)row * W;
  for (int i = 0; i < per; ++i) {
    const int d = lane + (i << 5);
    float o = (v[i] - mu) * inv * gam[d] + bet[d];
    if (do_tanh) o = tanhf(o);
    yr[d] = o;
  }
}

// RPE -> per-head coordinate scales sqw[h][c] = sqrt(2 * nqw[h][c]).
__global__ __launch_bounds__(64)
void rpe_prep_k(const float* __restrict__ rpe, float* __restrict__ sqw)
{
  __shared__ float es[64];
  __shared__ float qw[16];
  const int t = threadIdx.x;          // t = h*8 + cc*4 + nw
  const int h = t >> 3, cc = (t >> 2) & 1, nw = t & 3;
  float s = 0.f;
  for (int d = 0; d < 128; ++d) s += rpe[(h * 128 + d) * 8 + cc * 4 + nw];
  es[t] = __expf(fminf(s, 50.f));
  __syncthreads();
  if (nw == 0) qw[h * 2 + cc] = es[t] + es[t + 1] + es[t + 2] + es[t + 3];
  __syncthreads();
  if (t < 24) {
    const int hh = t / 3, c = t % 3;
    const int cc2 = (c == 0) ? 0 : (c - 1);
    sqw[t] = sqrtf(2.f * qw[hh * 2 + cc2]);
  }
}

// Fill coordinate columns 128..130 of q/k (width 132), zero the pad column.
__global__ void coordfill_k(float* __restrict__ q, float* __restrict__ k,
                            const float* __restrict__ coords,
                            const float* __restrict__ sqw, int h, int n)
{
  const int i = blockIdx.x * blockDim.x + threadIdx.x;
  if (i >= n) return;
  float* qr = q + (size_t)i * KQ;
  float* kr = k + (size_t)i * KQ;
  for (int c = 0; c < 3; ++c) {
    const float v = sqw[h * 3 + c] * coords[i * 3 + c];
    qr[128 + c] = v; kr[128 + c] = v;
  }
  qr[131] = 0.f; kr[131] = 0.f;
}

// -0.5 * ||row||^2 for q and k (pad column is zero so K=132 sum is fine).
__global__ void rowsq_k(const float* __restrict__ q, const float* __restrict__ k,
                        float* __restrict__ qs, float* __restrict__ ks, int n)
{
  const int i = blockIdx.x * blockDim.x + threadIdx.x;
  if (i >= n) return;
  float sq = 0.f, sk = 0.f;
  for (int d = 0; d < KQ; ++d) {
    const float a = q[(size_t)i * KQ + d]; sq += a * a;
    const float b = k[(size_t)i * KQ + d]; sk += b * b;
  }
  qs[i] = -0.5f * sq;
  ks[i] = -0.5f * sk;
}

// dn[row] = 1 / (sum_j P[row][j] + 1e-20); one wave per 128-wide row.
__global__ __launch_bounds__(256)
void rowsuminv_k(const float* __restrict__ p, float* __restrict__ dn)
{
  const int row  = blockIdx.x * 8 + (threadIdx.x >> 5);
  const int lane = threadIdx.x & 31;
  const float* pr = p + (size_t)row * BLKA;
  float s = pr[lane] + pr[lane + 32] + pr[lane + 64] + pr[lane + 96];
  for (int off = 16; off; off >>= 1) s += __shfl_xor(s, off, 32);
  dn[row] = 1.f / (s + 1e-20f);
}

// xout = xin + t + ol_b[d]  (elementwise over N*128)
__global__ void addbias_res_k(const float* __restrict__ xin,
                              const float* __restrict__ t,
                              const float* __restrict__ b,
                              float* __restrict__ xout, int n)
{
  const int i = blockIdx.x * blockDim.x + threadIdx.x;
  if (i >= n) return;
  xout[i] = xin[i] + t[i] + b[i & 127];
}

// Deterministic two-stage masked mean over gathered rows of z (width 64).
__global__ __launch_bounds__(256)
void meanpart_k(const float* __restrict__ z, const int* __restrict__ seq,
                int cnt, float* __restrict__ partial)
{
  __shared__ float red[256];
  const int p = blockIdx.x;           // 64 blocks
  const int tid = threadIdx.x;
  const int d = tid & 63, r = tid >> 6;
  const int chunk = (cnt + 63) / 64;
  const int t0 = p * chunk;
  int t1 = t0 + chunk; if (t1 > cnt) t1 = cnt;
  float s = 0.f;
  for (int t = t0 + r; t < t1; t += 4) s += z[(size_t)seq[t] * 64 + d];
  red[tid] = s;
  __syncthreads();
  if (r == 0) partial[p * 64 + d] = red[d] + red[64 + d] + red[128 + d] + red[192 + d];
}

__global__ __launch_bounds__(64)
void meanfinal_k(const float* __restrict__ partial, float* __restrict__ pooled, int cnt)
{
  const int d = threadIdx.x;
  float s = 0.f;
  for (int p = 0; p < 64; ++p) s += partial[p * 64 + d];
  pooled[d] = s / (float)cnt;
}

// out[o] = op_b[o] + pooled . op_w[:,o]
__global__ __launch_bounds__(32)
void head_k(const float* __restrict__ pooled, const float* __restrict__ w,
            const float* __restrict__ b, float* __restrict__ out)
{
  const int o = threadIdx.x;
  if (o >= 10) return;
  float s = b[o];
  for (int d = 0; d < 64; ++d) s += pooled[d] * w[d * 10 + o];
  out[o] = s;
}

// ---------------------------------------------------------------------------

extern "C" void kernel_launch(void* const* d_in, const int* in_sizes, int n_in,
                              void* d_out, int out_size, void* d_ws, size_t ws_size,
                              hipStream_t stream)
{
  (void)n_in; (void)out_size; (void)ws_size;
  const float* x_in   = (const float*)d_in[0];
  const float* coords = (const float*)d_in[1];
  const int*   seq    = (const int*)d_in[2];
  const int    cnt    = in_sizes[2];
  #define PRM(i) ((const float*)d_in[i])

  // ---- workspace layout (floats), ~355 MB total ----
  float* w = (float*)d_ws;
  size_t o = 0;
  auto alloc = [&](size_t nelem) { float* p = w + o; o += nelem; return p; };
  float* X0 = alloc((size_t)NP * 128);   // x after feat encoder
  float* X1 = alloc((size_t)NP * 128);   // x after block 1
  float* X2 = alloc((size_t)NP * 128);   // x after block 2
  float* T1 = alloc((size_t)NP * 128);   // layernorm output (xn)
  float* T2 = alloc((size_t)NP * 128);   // attn accumulator / ff hidden
  float* QB = alloc((size_t)NP * KQ);    // q (head-local, padded width 132)
  float* KB = alloc((size_t)NP * KQ);    // k
  float* VB = alloc((size_t)NP * 128);   // v
  float* PB = alloc((size_t)NP * 128);   // exp-scores (block-local)
  float* AB = alloc((size_t)NP * 128);   // aggregated per-head output
  float* QS = alloc((size_t)NP);         // -0.5||q||^2
  float* KS = alloc((size_t)NP);         // -0.5||k||^2
  float* DN = alloc((size_t)NP);         // 1/denominator
  float* Z  = alloc((size_t)NP * 64);    // tanh(all_x @ W_w)
  float* SQW  = alloc(32);               // sqrt(2*nqw) per (head,coord)
  float* PART = alloc(64 * 64);
  float* POOL = alloc(64);
  // MLP hidden buffers alias the (now idle) attention scratch:
  float* H1 = QB;                        // needs NP*256 <= NP*264 (QB..KB)  OK
  float* H2 = VB;                        // needs NP*256 == NP*256 (VB..PB)  OK

  const float* NUL = nullptr;
  const dim3 TPB(128);
  #define GRID(NC) dim3((NC) / 64, NP / 128)

  // ---- feature encoder: relu(x@W1+b1)@W2+b2 ----
  gemm_wmma<0,0,0,EPI_RELU><<<GRID(128), TPB, 0, stream>>>(
      x_in, NUL, NUL, INW, PRM(3), 128, PRM(4), T1, 128, INW, NUL, NUL, NUL, 0);
  gemm_wmma<0,0,0,EPI_NONE><<<GRID(128), TPB, 0, stream>>>(
      T1, NUL, NUL, 128, PRM(5), 128, PRM(6), X0, 128, 128, NUL, NUL, NUL, 0);

  // ---- two transformer blocks ----
  for (int L = 0; L < 2; ++L) {
    const int bi = 7 + L * 14;
    const float* Xin  = L ? X1 : X0;
    float*       Xout = L ? X2 : X1;
    const float* wq  = PRM(bi + 4);
    const float* wk  = PRM(bi + 5);
    const float* wv  = PRM(bi + 6);
    const float* olw = PRM(bi + 8);
    const float* olb = PRM(bi + 9);

    layernorm_k<<<dim3(NP / 8), dim3(256), 0, stream>>>(
        Xin, PRM(bi + 0), PRM(bi + 1), T1, 128, 0);
    rpe_prep_k<<<dim3(1), dim3(64), 0, stream>>>(PRM(bi + 7), SQW);

    for (int h = 0; h < NH; ++h) {
      // per-head projections: xn @ wq/wk/wv[:, h*128:(h+1)*128]
      gemm_wmma<0,0,0,EPI_NONE><<<GRID(128), TPB, 0, stream>>>(
          T1, NUL, NUL, 128, wq + h * 128, NH * DH, NUL, QB, KQ, 128,
          NUL, NUL, NUL, 0);
      gemm_wmma<0,0,0,EPI_NONE><<<GRID(128), TPB, 0, stream>>>(
          T1, NUL, NUL, 128, wk + h * 128, NH * DH, NUL, KB, KQ, 128,
          NUL, NUL, NUL, 0);
      gemm_wmma<0,0,0,EPI_NONE><<<GRID(128), TPB, 0, stream>>>(
          T1, NUL, NUL, 128, wv + h * 128, NH * DH, NUL, VB, 128, 128,
          NUL, NUL, NUL, 0);
      coordfill_k<<<dim3(NP / 256), dim3(256), 0, stream>>>(
          QB, KB, coords, SQW, h, NP);
      rowsq_k<<<dim3(NP / 256), dim3(256), 0, stream>>>(QB, KB, QS, KS, NP);
      // block-diagonal Q @ K^T with Gaussian-kernel epilogue
      gemm_wmma<0,1,1,EPI_EXPSCORE><<<GRID(128), TPB, 0, stream>>>(
          QB, NUL, NUL, KQ, KB, KQ, NUL, PB, 128, KQ, QS, KS, NUL, 0);
      rowsuminv_k<<<dim3(NP / 8), dim3(256), 0, stream>>>(PB, DN);
      // block-diagonal P @ V, scaled by 1/denominator
      gemm_wmma<0,0,1,EPI_SCALE_ROW><<<GRID(128), TPB, 0, stream>>>(
          PB, NUL, NUL, 128, VB, 128, NUL, AB, 128, 128, DN, NUL, NUL, 0);
      // accumulate per-head slice of output projection
      if (h == 0)
        gemm_wmma<0,0,0,EPI_NONE><<<GRID(128), TPB, 0, stream>>>(
            AB, NUL, NUL, 128, olw, 128, NUL, T2, 128, 128, NUL, NUL, NUL, 0);
      else
        gemm_wmma<0,0,0,EPI_ACC><<<GRID(128), TPB, 0, stream>>>(
            AB, NUL, NUL, 128, olw + (size_t)h * 128 * 128, 128, NUL, T2, 128,
            128, NUL, NUL, NUL, 0);
    }
    addbias_res_k<<<dim3(NP * 128 / 256), dim3(256), 0, stream>>>(
        Xin, T2, olb, Xout, NP * 128);

    // feed-forward with pre-LN and residual
    layernorm_k<<<dim3(NP / 8), dim3(256), 0, stream>>>(
        Xout, PRM(bi + 2), PRM(bi + 3), T1, 128, 0);
    gemm_wmma<0,0,0,EPI_RELU><<<GRID(128), TPB, 0, stream>>>(
        T1, NUL, NUL, 128, PRM(bi + 10), 128, PRM(bi + 11), T2, 128, 128,
        NUL, NUL, NUL, 0);
    gemm_wmma<0,0,0,EPI_RES><<<GRID(128), TPB, 0, stream>>>(
        T2, NUL, NUL, 128, PRM(bi + 12), 128, PRM(bi + 13), Xout, 128, 128,
        NUL, NUL, Xout, 128);
  }

  // ---- z = tanh([X0|X1|X2] @ W_w) ----
  gemm_wmma<1,0,0,EPI_TANH><<<GRID(64), TPB, 0, stream>>>(
      X0, X1, X2, 128, PRM(35), 64, NUL, Z, 64, 384, NUL, NUL, NUL, 0);

  // ---- MLP: 64->256 ->256 ->256 ->256 ->64, tanh(LN(.)) between ----
  gemm_wmma<0,0,0,EPI_NONE><<<GRID(256), TPB, 0, stream>>>(
      Z, NUL, NUL, 64, PRM(36), 256, PRM(37), H1, 256, 64, NUL, NUL, NUL, 0);
  layernorm_k<<<dim3(NP / 8), dim3(256), 0, stream>>>(H1, PRM(38), PRM(39), H1, 256, 1);
  gemm_wmma<0,0,0,EPI_NONE><<<GRID(256), TPB, 0, stream>>>(
      H1, NUL, NUL, 256, PRM(40), 256, PRM(41), H2, 256, 256, NUL, NUL, NUL, 0);
  layernorm_k<<<dim3(NP / 8), dim3(256), 0, stream>>>(H2, PRM(42), PRM(43), H2, 256, 1);
  gemm_wmma<0,0,0,EPI_NONE><<<GRID(256), TPB, 0, stream>>>(
      H2, NUL, NUL, 256, PRM(44), 256, PRM(45), H1, 256, 256, NUL, NUL, NUL, 0);
  layernorm_k<<<dim3(NP / 8), dim3(256), 0, stream>>>(H1, PRM(46), PRM(47), H1, 256, 1);
  gemm_wmma<0,0,0,EPI_NONE><<<GRID(256), TPB, 0, stream>>>(
      H1, NUL, NUL, 256, PRM(48), 256, PRM(49), H2, 256, 256, NUL, NUL, NUL, 0);
  layernorm_k<<<dim3(NP / 8), dim3(256), 0, stream>>>(H2, PRM(50), PRM(51), H2, 256, 1);
  // z = z + (h @ m_w4 + m_b4)
  gemm_wmma<0,0,0,EPI_RES><<<GRID(64), TPB, 0, stream>>>(
      H2, NUL, NUL, 256, PRM(52), 64, PRM(53), Z, 64, 256, NUL, NUL, Z, 64);

  // ---- masked mean + output head ----
  meanpart_k<<<dim3(64), dim3(256), 0, stream>>>(Z, seq, cnt, PART);
  meanfinal_k<<<dim3(1), dim3(64), 0, stream>>>(PART, POOL, cnt);
  head_k<<<dim3(1), dim3(32), 0, stream>>>(POOL, PRM(54), PRM(55), (float*)d_out);
  #undef PRM
  #undef GRID
}